// GAT_14989435863258
// MI455X (gfx1250) — compile-verified
//
#include <hip/hip_runtime.h>
#include <hip/hip_bf16.h>
#include <stdint.h>

// ---------------------------------------------------------------------------
// GATv2 (P=2, H=2, F=64) + JK-cat + global-attention pool + MLP, for gfx1250.
// fp32 everywhere; node linear transforms use V_WMMA_F32_16X16X4_F32.
// ---------------------------------------------------------------------------

typedef __attribute__((ext_vector_type(2))) float v2f;
typedef __attribute__((ext_vector_type(8))) float v8f;

#define FDIM 64
#define HF   128   // H*F

__device__ __forceinline__ float lrelu(float x, float s) { return x > 0.0f ? x : s * x; }

// order-preserving float<->int map (valid for all non-NaN floats incl. +-inf)
__device__ __forceinline__ int ord_from_f32(float f) {
  int i = __float_as_int(f);
  return i >= 0 ? i : (i ^ 0x7fffffff);
}
__device__ __forceinline__ float f32_from_ord(int i) {
  return __int_as_float(i >= 0 ? i : (i ^ 0x7fffffff));
}

// ------------------------------- fills -------------------------------------
__global__ void fill_f32_kernel(float* __restrict__ p, float v, long long n) {
  long long t = (long long)blockIdx.x * blockDim.x + threadIdx.x;
  if (t < n) p[t] = v;
}
__global__ void fill_i32_kernel(int* __restrict__ p, int v, long long n) {
  long long t = (long long)blockIdx.x * blockDim.x + threadIdx.x;
  if (t < n) p[t] = v;
}

// ----------------------- fused node transform (WMMA) -----------------------
// XL = X @ Wl + bl ; XR = X @ Wr + br      X:[n,64]  W:[64,128]  out:[n,128]
// block = 256 threads = 8 waves; block covers 16 rows, wave w covers cols 16w..16w+15
__global__ void __launch_bounds__(256)
gat_node_transform(const float* __restrict__ X,
                   const float* __restrict__ Wl, const float* __restrict__ bl,
                   const float* __restrict__ Wr, const float* __restrict__ br,
                   float* __restrict__ XL, float* __restrict__ XR, int nrows) {
  const int lane = threadIdx.x & 31;
  const int wave = threadIdx.x >> 5;        // 0..7 -> column tile
  const int row0 = blockIdx.x * 16;
  const int col0 = wave * 16;
  const int half = lane >> 4;               // 0: lanes 0-15, 1: lanes 16-31
  const int l16  = lane & 15;

  // A operand row for this lane (clamped for tail safety; stores predicated)
  int arow_i = row0 + l16;
  if (arow_i >= nrows) arow_i = nrows - 1;
  const float* arow = X + (size_t)arow_i * FDIM;

  v8f cL = {}; v8f cR = {};
#pragma unroll
  for (int k0 = 0; k0 < FDIM; k0 += 4) {
    const int ka = k0 + 2 * half;           // even -> 8B aligned
    v2f a;
    {
      const float2 a2 = *(const float2*)(arow + ka);
      a.x = a2.x; a.y = a2.y;
    }
    v2f bL, bR;
    bL.x = Wl[(size_t)ka * HF + col0 + l16];
    bL.y = Wl[(size_t)(ka + 1) * HF + col0 + l16];
    bR.x = Wr[(size_t)ka * HF + col0 + l16];
    bR.y = Wr[(size_t)(ka + 1) * HF + col0 + l16];
    cL = __builtin_amdgcn_wmma_f32_16x16x4_f32(false, a, false, bL, (short)0, cL, false, false);
    cR = __builtin_amdgcn_wmma_f32_16x16x4_f32(false, a, false, bR, (short)0, cR, false, false);
  }

  const int col = col0 + l16;
  const float biasL = bl[col];
  const float biasR = br[col];
#pragma unroll
  for (int i = 0; i < 8; ++i) {
    const int row = row0 + i + 8 * half;    // D layout: VGPR i holds rows i / i+8
    if (row < nrows) {
      XL[(size_t)row * HF + col] = cL[i] + biasL;
      XR[(size_t)row * HF + col] = cR[i] + biasR;
    }
  }
}

// --------------------------- edge pass A: logits ---------------------------
// one wave per edge; lane covers 4 of the 128 (h,f) slots; half-wave = head
__global__ void __launch_bounds__(256)
gat_edge_logits(const int* __restrict__ ei, const float* __restrict__ ea,
                const float* __restrict__ XL, const float* __restrict__ XR,
                const float* __restrict__ We,   // [128]
                const float* __restrict__ att,  // [2*64] flat == f-index order
                float* __restrict__ logits, int* __restrict__ lmax_i, int E) {
  const int e = (int)(((long long)blockIdx.x * blockDim.x + threadIdx.x) >> 5);
  if (e >= E) return;                       // uniform per wave
  const int lane = threadIdx.x & 31;
  const int src = ei[e];
  const int dst = ei[E + e];
  const float eav = ea[e];
  const int f0 = lane * 4;

  const float4 xl = *(const float4*)(XL + (size_t)src * HF + f0);
  const float4 xr = *(const float4*)(XR + (size_t)dst * HF + f0);
  const float4 we = *(const float4*)(We + f0);
  const float4 at = *(const float4*)(att + f0);

  float s = 0.0f;
  {
    float m;
    m = xl.x + xr.x + eav * we.x; s += lrelu(m, 0.2f) * at.x;
    m = xl.y + xr.y + eav * we.y; s += lrelu(m, 0.2f) * at.y;
    m = xl.z + xr.z + eav * we.z; s += lrelu(m, 0.2f) * at.z;
    m = xl.w + xr.w + eav * we.w; s += lrelu(m, 0.2f) * at.w;
  }
  // reduce within each 16-lane half (one head per half)
  s += __shfl_xor(s, 8, 32);
  s += __shfl_xor(s, 4, 32);
  s += __shfl_xor(s, 2, 32);
  s += __shfl_xor(s, 1, 32);

  if ((lane & 15) == 0) {
    const int h = lane >> 4;
    logits[(size_t)e * 2 + h] = s;
    atomicMax(&lmax_i[dst * 2 + h], ord_from_f32(s));
  }
}

// -------------------- edge pass B: exp + denom (segment sum) ---------------
__global__ void gat_edge_expsum(const int* __restrict__ ei, float* __restrict__ logits,
                                const int* __restrict__ lmax_i, float* __restrict__ denom,
                                int E) {
  const long long t = (long long)blockIdx.x * blockDim.x + threadIdx.x;
  if (t >= 2LL * E) return;
  const int e = (int)(t >> 1), h = (int)(t & 1);
  const int dst = ei[E + e];
  const float lm = f32_from_ord(lmax_i[dst * 2 + h]);
  const float a = __expf(logits[t] - lm);
  logits[t] = a;                             // reuse buffer for 'a'
  atomicAdd(&denom[dst * 2 + h], a);
}

// -------------------- edge pass C: weighted scatter-add --------------------
__global__ void __launch_bounds__(256)
gat_edge_scatter(const int* __restrict__ ei, const float* __restrict__ XL,
                 const float* __restrict__ aexp, const float* __restrict__ denom,
                 float* __restrict__ accum, int E) {
  const int e = (int)(((long long)blockIdx.x * blockDim.x + threadIdx.x) >> 5);
  if (e >= E) return;
  const int lane = threadIdx.x & 31;
  const int src = ei[e];
  const int dst = ei[E + e];
  const int h = lane >> 4;
  const float alpha = aexp[(size_t)e * 2 + h] / denom[dst * 2 + h];
  const int f0 = lane * 4;
  const float4 xl = *(const float4*)(XL + (size_t)src * HF + f0);
  float* ap = accum + (size_t)dst * HF + f0;
  atomicAdd(ap + 0, xl.x * alpha);
  atomicAdd(ap + 1, xl.y * alpha);
  atomicAdd(ap + 2, xl.z * alpha);
  atomicAdd(ap + 3, xl.w * alpha);
}

// --------------------- head mean + bias + leaky_relu -----------------------
__global__ void gat_finalize(const float* __restrict__ accum, const float* __restrict__ gbias,
                             float* __restrict__ hout, int n) {
  const long long t = (long long)blockIdx.x * blockDim.x + threadIdx.x;
  if (t >= (long long)n * FDIM) return;
  const int node = (int)(t >> 6), f = (int)(t & 63);
  const float v = 0.5f * (accum[(size_t)node * HF + f] + accum[(size_t)node * HF + FDIM + f])
                + gbias[f];
  hout[t] = lrelu(v, 0.01f);
}

// ------------------------- global attention gate ---------------------------
// one wave per node; dot over 192 dims (x|h1|h2), 6 elems/lane
__global__ void __launch_bounds__(256)
gate_kernel(const float* __restrict__ x, const float* __restrict__ h1,
            const float* __restrict__ h2, const float* __restrict__ gw,
            const float* __restrict__ gb, float* __restrict__ gate,
            int* __restrict__ gmax_i, int n) {
  const int node = (int)(((long long)blockIdx.x * blockDim.x + threadIdx.x) >> 5);
  if (node >= n) return;
  const int lane = threadIdx.x & 31;
  const size_t b = (size_t)node * FDIM;
  float s = x[b + lane] * gw[lane]        + x[b + 32 + lane] * gw[32 + lane]
          + h1[b + lane] * gw[64 + lane]  + h1[b + 32 + lane] * gw[96 + lane]
          + h2[b + lane] * gw[128 + lane] + h2[b + 32 + lane] * gw[160 + lane];
  s += __shfl_xor(s, 16, 32);
  s += __shfl_xor(s, 8, 32);
  s += __shfl_xor(s, 4, 32);
  s += __shfl_xor(s, 2, 32);
  s += __shfl_xor(s, 1, 32);
  if (lane == 0) {
    s = lrelu(s + gb[0], 0.01f);
    gate[node] = s;
    atomicMax(gmax_i, ord_from_f32(s));
  }
}

// ---------------- pooled numerator S[192] and denominator Z ----------------
__global__ void __launch_bounds__(256)
pool_kernel(const float* __restrict__ x, const float* __restrict__ h1,
            const float* __restrict__ h2, const float* __restrict__ gate,
            const int* __restrict__ gmax_i, float* __restrict__ S,
            float* __restrict__ Z, int n) {
  const int lane = threadIdx.x & 31;
  const int wid = (int)(((long long)blockIdx.x * blockDim.x + threadIdx.x) >> 5);
  const int nw = (int)(((long long)gridDim.x * blockDim.x) >> 5);
  const float gm = f32_from_ord(*gmax_i);
  float a0 = 0, a1 = 0, a2 = 0, a3 = 0, a4 = 0, a5 = 0, zacc = 0;
  for (int node = wid; node < n; node += nw) {
    const float ev = __expf(gate[node] - gm);
    const size_t b = (size_t)node * FDIM;
    a0 += ev * x[b + lane];   a1 += ev * x[b + 32 + lane];
    a2 += ev * h1[b + lane];  a3 += ev * h1[b + 32 + lane];
    a4 += ev * h2[b + lane];  a5 += ev * h2[b + 32 + lane];
    zacc += ev;
  }
  atomicAdd(&S[lane],       a0);
  atomicAdd(&S[32 + lane],  a1);
  atomicAdd(&S[64 + lane],  a2);
  atomicAdd(&S[96 + lane],  a3);
  atomicAdd(&S[128 + lane], a4);
  atomicAdd(&S[160 + lane], a5);
  if (lane == 0) atomicAdd(Z, zacc);   // zacc identical across lanes
}

// ------------------------------ final MLP ----------------------------------
// z = [S/Z (192), problemType] (193) -> 96 -> 96 -> 2
__global__ void __launch_bounds__(256)
mlp_kernel(const float* __restrict__ S, const float* __restrict__ Z,
           const float* __restrict__ pt,
           const float* __restrict__ w1, const float* __restrict__ b1,
           const float* __restrict__ w2, const float* __restrict__ b2,
           const float* __restrict__ w3, const float* __restrict__ b3,
           float* __restrict__ out) {
  __shared__ float zin[193];
  __shared__ float z1[96];
  __shared__ float z2[96];
  const int t = threadIdx.x;
  const float invZ = 1.0f / Z[0];
  if (t < 192) zin[t] = S[t] * invZ;
  if (t == 0) zin[192] = pt[0];
  __syncthreads();
  if (t < 96) {
    float s = b1[t];
    for (int k = 0; k < 193; ++k) s += zin[k] * w1[(size_t)k * 96 + t];
    z1[t] = lrelu(s, 0.01f);
  }
  __syncthreads();
  if (t < 96) {
    float s = b2[t];
    for (int k = 0; k < 96; ++k) s += z1[k] * w2[(size_t)k * 96 + t];
    z2[t] = lrelu(s, 0.01f);
  }
  __syncthreads();
  if (t < 2) {
    float s = b3[t];
    for (int k = 0; k < 96; ++k) s += z2[k] * w3[(size_t)k * 2 + t];
    out[t] = s;
  }
}

// ---------------------------------------------------------------------------
extern "C" void kernel_launch(void* const* d_in, const int* in_sizes, int n_in,
                              void* d_out, int out_size, void* d_ws, size_t ws_size,
                              hipStream_t stream) {
  (void)n_in; (void)out_size; (void)ws_size;
  const float* x        = (const float*)d_in[0];   // [N,64]
  const int*   ei       = (const int*)  d_in[1];   // [2,E]
  const float* ea       = (const float*)d_in[2];   // [E,1]
  const float* pt       = (const float*)d_in[3];   // [1]
  const float* lin_l_w  = (const float*)d_in[5];   // [2,64,128]
  const float* lin_l_b  = (const float*)d_in[6];   // [2,128]
  const float* lin_r_w  = (const float*)d_in[7];   // [2,64,128]
  const float* lin_r_b  = (const float*)d_in[8];   // [2,128]
  const float* lin_e_w  = (const float*)d_in[9];   // [2,1,128]
  const float* att      = (const float*)d_in[10];  // [2,2,64]
  const float* gat_bias = (const float*)d_in[11];  // [2,64]
  const float* gate_w   = (const float*)d_in[12];  // [192,1]
  const float* gate_b   = (const float*)d_in[13];  // [1]
  const float* fc1_w    = (const float*)d_in[14];  // [193,96]
  const float* fc1_b    = (const float*)d_in[15];
  const float* fc2_w    = (const float*)d_in[16];  // [96,96]
  const float* fc2_b    = (const float*)d_in[17];
  const float* fcl_w    = (const float*)d_in[18];  // [96,2]
  const float* fcl_b    = (const float*)d_in[19];

  const int N = in_sizes[0] / FDIM;
  const int E = in_sizes[1] / 2;

  // ---- workspace layout (floats) ----
  float* ws = (float*)d_ws;
  size_t off = 0;
  float* xl     = ws + off; off += (size_t)N * HF;
  float* xr     = ws + off; off += (size_t)N * HF;   // reused as 'accum' per layer
  float* logits = ws + off; off += (size_t)E * 2;
  int*   lmax_i = (int*)(ws + off); off += (size_t)N * 2;
  float* denom  = ws + off; off += (size_t)N * 2;
  float* h1     = ws + off; off += (size_t)N * FDIM;
  float* h2     = ws + off; off += (size_t)N * FDIM;
  float* gate   = ws + off; off += (size_t)N;
  float* S      = ws + off; off += 192;
  float* Z      = ws + off; off += 1;
  int*   gmax_i = (int*)(ws + off); off += 1;

  const int blkRows   = (N + 15) / 16;
  const int blkEdgeW  = (E + 7) / 8;                     // wave-per-edge kernels
  const long long nEH = 2LL * E;

  for (int p = 0; p < 2; ++p) {
    const float* hin  = (p == 0) ? x  : h1;
    float*       hout = (p == 0) ? h1 : h2;
    const size_t wofs = (size_t)p * FDIM * HF;

    gat_node_transform<<<blkRows, 256, 0, stream>>>(
        hin, lin_l_w + wofs, lin_l_b + (size_t)p * HF,
             lin_r_w + wofs, lin_r_b + (size_t)p * HF, xl, xr, N);

    fill_i32_kernel<<<(int)((2LL * N + 255) / 256), 256, 0, stream>>>(
        lmax_i, (int)0x80000000, 2LL * N);
    fill_f32_kernel<<<(int)((2LL * N + 255) / 256), 256, 0, stream>>>(
        denom, 0.0f, 2LL * N);

    gat_edge_logits<<<blkEdgeW, 256, 0, stream>>>(
        ei, ea, xl, xr, lin_e_w + (size_t)p * HF, att + (size_t)p * HF,
        logits, lmax_i, E);

    // xr no longer needed -> becomes the message accumulator
    fill_f32_kernel<<<(int)(((long long)N * HF + 255) / 256), 256, 0, stream>>>(
        xr, 0.0f, (long long)N * HF);

    gat_edge_expsum<<<(int)((nEH + 255) / 256), 256, 0, stream>>>(
        ei, logits, lmax_i, denom, E);

    gat_edge_scatter<<<blkEdgeW, 256, 0, stream>>>(ei, xl, logits, denom, xr, E);

    gat_finalize<<<(int)(((long long)N * FDIM + 255) / 256), 256, 0, stream>>>(
        xr, gat_bias + (size_t)p * FDIM, hout, N);
  }

  // ---- global attention pooling ----
  fill_i32_kernel<<<1, 32, 0, stream>>>(gmax_i, (int)0x80000000, 1);
  gate_kernel<<<(int)(((long long)N * 32 + 255) / 256), 256, 0, stream>>>(
      x, h1, h2, gate_w, gate_b, gate, gmax_i, N);
  fill_f32_kernel<<<1, 256, 0, stream>>>(S, 0.0f, 193);   // zero S[192] + Z[1]
  pool_kernel<<<256, 256, 0, stream>>>(x, h1, h2, gate, gmax_i, S, Z, N);

  // ---- final MLP ----
  mlp_kernel<<<1, 256, 0, stream>>>(S, Z, pt, fc1_w, fc1_b, fc2_w, fc2_b,
                                    fcl_w, fcl_b, (float*)d_out);
}